// NMS_64879775973772
// MI455X (gfx1250) — compile-verified
//
#include <hip/hip_runtime.h>
#include <hip/hip_bf16.h>

#define NBOX      8192
#define SORT_T    1024
#define TILE      256
#define IOU_THR   0.7f

// ---------------------------------------------------------------------------
// Kernel 1: single-workgroup bitonic sort of (score, index) keys in LDS,
// then gather boxes into sorted order in the workspace.
// Key = (score_bits << 32) | (0xFFFFFFFF - index); descending sort of key
// == stable argsort(-score). We store ~key and sort ascending.
// ---------------------------------------------------------------------------
__global__ void __launch_bounds__(SORT_T)
nms_sort_kernel(const float4* __restrict__ boxes,
                const float*  __restrict__ score,
                float4* __restrict__ sorted_boxes) {
    __shared__ unsigned long long keys[NBOX];   // 64 KB (WGP has 320 KB)
    const int tid = threadIdx.x;

    for (int i = tid; i < NBOX; i += SORT_T) {
        unsigned sb = __float_as_uint(score[i]);            // scores in [0,1): monotonic bits
        unsigned long long k =
            ((unsigned long long)sb << 32) | (unsigned)(0xFFFFFFFFu - (unsigned)i);
        keys[i] = ~k;                                        // ascending(~k) == descending(k)
    }
    __syncthreads();

    for (int k = 2; k <= NBOX; k <<= 1) {
        for (int j = k >> 1; j > 0; j >>= 1) {
            for (int i = tid; i < NBOX; i += SORT_T) {
                int ixj = i ^ j;
                if (ixj > i) {
                    unsigned long long a = keys[i];
                    unsigned long long b = keys[ixj];
                    bool up = ((i & k) == 0);
                    if ((a > b) == up) { keys[i] = b; keys[ixj] = a; }
                }
            }
            __syncthreads();
        }
    }

    for (int i = tid; i < NBOX; i += SORT_T) {
        unsigned long long orig = ~keys[i];
        unsigned idx = 0xFFFFFFFFu - (unsigned)(orig & 0xFFFFFFFFull);
        sorted_boxes[i] = boxes[idx];
    }
}

// ---------------------------------------------------------------------------
// Kernel 2: suppression. Thread j keeps its sorted box in registers; i-tiles
// of 256 boxes are staged into LDS with the CDNA5 async data mover
// (global_load_async_to_lds_b128, tracked by ASYNCcnt), then scanned.
// keep[j] = !any(i < j : IoU(i,j) > 0.7). Out = box * keep.
// ---------------------------------------------------------------------------
__global__ void __launch_bounds__(TILE)
nms_suppress_kernel(const float4* __restrict__ sorted_boxes,
                    float4* __restrict__ out) {
#pragma clang fp contract(off)
    __shared__ float4 tile[TILE];               // 4 KB tile of i-boxes
    const int tid = threadIdx.x;
    const int blk = blockIdx.x;
    const int j   = blk * TILE + tid;

    // boxes are [y1, x1, y2, x2] -> float4 {x=y1, y=x1, z=y2, w=x2}
    const float4 bj = sorted_boxes[j];
    const float area_j = (bj.z - bj.x) * (bj.w - bj.y);

    // LDS byte offset of this lane's tile slot: flat LDS addr low 32 bits
    // are the workgroup-relative LDS offset (ISA aperture model).
    unsigned lds_off = (unsigned)(unsigned long long)(&tile[tid]);

    bool supp = false;

    for (int t = 0; t <= blk; ++t) {
        // Async DMA: each lane pulls one 16B box into its LDS slot.
        unsigned long long gaddr =
            (unsigned long long)(const void*)(sorted_boxes + (t * TILE + tid));
        asm volatile("global_load_async_to_lds_b128 %0, %1, off"
                     :: "v"(lds_off), "v"(gaddr) : "memory");
        asm volatile("s_wait_asynccnt 0" ::: "memory");
        __syncthreads();                        // all waves' tile data visible

        const int limit = min(TILE, j - t * TILE);   // enforce i < j (strict triu)
        for (int kk = 0; kk < limit; ++kk) {
            float4 bi = tile[kk];
            // exact op sequence of the reference (_pairwise_iou)
            float iy1 = fmaxf(bj.x, bi.x);
            float ix1 = fmaxf(bj.y, bi.y);
            float iy2 = fminf(bj.z, bi.z);
            float ix2 = fminf(bj.w, bi.w);
            float ih  = fmaxf(iy2 - iy1, 0.0f);
            float iw  = fmaxf(ix2 - ix1, 0.0f);
            float inter  = ih * iw;
            float area_i = (bi.z - bi.x) * (bi.w - bi.y);
            float uni    = area_i + area_j - inter;
            float iou    = inter / fmaxf(uni, 1e-9f);
            supp = supp || (iou > IOU_THR);
        }
        __syncthreads();                        // protect tile WAR before next DMA
    }

    const float s = supp ? 0.0f : 1.0f;
    out[j] = make_float4(bj.x * s, bj.y * s, bj.z * s, bj.w * s);
}

// ---------------------------------------------------------------------------
extern "C" void kernel_launch(void* const* d_in, const int* in_sizes, int n_in,
                              void* d_out, int out_size, void* d_ws, size_t ws_size,
                              hipStream_t stream) {
    const float4* boxes = (const float4*)d_in[0];   // (8192, 4) f32
    const float*  score = (const float*)d_in[1];    // (8192,)   f32
    float4* sorted_boxes = (float4*)d_ws;           // 128 KB scratch
    float4* out = (float4*)d_out;                   // (8192, 4) f32

    nms_sort_kernel<<<1, SORT_T, 0, stream>>>(boxes, score, sorted_boxes);
    nms_suppress_kernel<<<NBOX / TILE, TILE, 0, stream>>>(sorted_boxes, out);
}